// CSPCompBlock_27805618274736
// MI455X (gfx1250) — compile-verified
//
#include <hip/hip_runtime.h>
#include <hip/hip_bf16.h>
#include <stdint.h>

typedef __attribute__((ext_vector_type(2))) float        v2f;
typedef __attribute__((ext_vector_type(8))) float        v8f;
typedef __attribute__((ext_vector_type(4))) unsigned int u32x4;
typedef __attribute__((ext_vector_type(4))) int          i32x4;
typedef __attribute__((ext_vector_type(8))) int          i32x8;

#define NB   8
#define CHN  128
#define DA   128
#define DB   128
#define EPSV 0.001f

#define TS     32              // output tile (a x b)
#define HALO   3               // receptive-field halo per side
#define TI     (TS + 2*HALO)   // 38
#define LP     40              // LDS row pitch (floats)

// ---------------------------------------------------------------------------
// Fused depthwise pseudo-block: conv(a,d1)+ReLU -> conv(b,d1)+ReLU ->
// conv(a,d2)+ReLU [WMMA] -> conv(b,d2) [WMMA] -> BN -> ReLU(residual).
// Input/output layout: (N=8, CH=128, A=128, B=128) contiguous.
// The dilation-2 convs are banded 16x20 x 20x16 matmuls on v_wmma_f32_16x16x4_f32.
// ---------------------------------------------------------------------------
__global__ __launch_bounds__(256)
void pseudo_stage(const float* __restrict__ in, float* __restrict__ out,
                  const float* __restrict__ w1, const float* __restrict__ b1,
                  const float* __restrict__ w2, const float* __restrict__ b2,
                  const float* __restrict__ w3, const float* __restrict__ b3,
                  const float* __restrict__ w4, const float* __restrict__ b4,
                  const float* __restrict__ g,  const float* __restrict__ bt,
                  const float* __restrict__ mm, const float* __restrict__ vv)
{
    __shared__ __attribute__((aligned(16))) float s0[TI * LP]; // 38x38 input + halo
    __shared__ __attribute__((aligned(16))) float s1[36 * LP]; // after conv1: 36x38
    __shared__ __attribute__((aligned(16))) float s2[36 * LP]; // after conv2: 36x36
    __shared__ __attribute__((aligned(16))) float s3[32 * LP]; // after conv3: 32x36

    const int blk = blockIdx.x;
    const int tb  = (blk & 3) * TS;
    const int ta  = ((blk >> 2) & 3) * TS;
    const int c   = (blk >> 4) & 127;
    const int n   = blk >> 11;

    const float* plane  = in  + (((size_t)n * CHN + c) * DA) * DB;
    float*       oplane = out + (((size_t)n * CHN + c) * DA) * DB;

    const float w1a = w1[3*c], w1b = w1[3*c+1], w1c = w1[3*c+2], bb1 = b1[c];
    const float w2a = w2[3*c], w2b = w2[3*c+1], w2c = w2[3*c+2], bb2 = b2[c];
    const float w3a = w3[3*c], w3b = w3[3*c+1], w3c = w3[3*c+2], bb3 = b3[c];
    const float w4a = w4[3*c], w4b = w4[3*c+1], w4c = w4[3*c+2], bb4 = b4[c];
    const float scale = g[c] * rsqrtf(vv[c] + EPSV);
    const float mu = mm[c], beta = bt[c];

    const int t = threadIdx.x;

    // ---- load 38x38 tile (zero-padded at tensor borders) -------------------
    for (int idx = t; idx < TI * TI; idx += 256) {
        int i = idx / TI, j = idx % TI;
        int a = ta - HALO + i, b = tb - HALO + j;
        float val = 0.f;
        if ((unsigned)a < (unsigned)DA && (unsigned)b < (unsigned)DB)
            val = plane[a * DB + b];
        s0[i * LP + j] = val;
    }
    __syncthreads();

    // ---- conv1 along a, dil 1, +ReLU: rows map a=A0-2+i, 36x38 -------------
    for (int idx = t; idx < 36 * TI; idx += 256) {
        int i = idx / TI, j = idx % TI;
        float r = fmaf(w1a, s0[i * LP + j],
                  fmaf(w1b, s0[(i + 1) * LP + j],
                  fmaf(w1c, s0[(i + 2) * LP + j], bb1)));
        s1[i * LP + j] = fmaxf(r, 0.f);
    }
    __syncthreads();

    // ---- conv2 along b, dil 1, +ReLU: cols map b=B0-2+j, 36x36 -------------
    for (int idx = t; idx < 36 * 36; idx += 256) {
        int i = idx / 36, j = idx % 36;
        float r = fmaf(w2a, s1[i * LP + j],
                  fmaf(w2b, s1[i * LP + j + 1],
                  fmaf(w2c, s1[i * LP + j + 2], bb2)));
        s2[i * LP + j] = fmaxf(r, 0.f);
    }
    __syncthreads();

    const int lane = t & 31;
    const int wave = t >> 5;
    const int mrow = lane & 15;   // M (row) / N (col) index within 16
    const int half = lane >> 4;   // lane half selects K pairs / row halves

    // ---- conv3 along a, dil 2, +ReLU (WMMA): out3 = K3(16x20) x out2(20x16)
    // K3[m][k] = w3[t] iff k == m + 2t. 6 tiles: rows {0,16} x cols {0,16,20}.
    if (wave < 6) {
        const int ri = (wave / 3) * 16;
        const int cjt[3] = {0, 16, 20};
        const int cj = cjt[wave % 3];
        v8f acc = {0.f,0.f,0.f,0.f,0.f,0.f,0.f,0.f};
        #pragma unroll
        for (int ck = 0; ck < 5; ++ck) {
            const int k0 = 4 * ck + 2 * half;   // this lane's two K indices: k0, k0+1
            v2f A, Bv;
            A.x = (k0     == mrow) ? w3a : (k0     == mrow + 2) ? w3b : (k0     == mrow + 4) ? w3c : 0.f;
            A.y = (k0 + 1 == mrow) ? w3a : (k0 + 1 == mrow + 2) ? w3b : (k0 + 1 == mrow + 4) ? w3c : 0.f;
            Bv.x = s2[(ri + k0)     * LP + cj + mrow];
            Bv.y = s2[(ri + k0 + 1) * LP + cj + mrow];
            acc = __builtin_amdgcn_wmma_f32_16x16x4_f32(false, A, false, Bv,
                                                        (short)0, acc, false, false);
        }
        #pragma unroll
        for (int r = 0; r < 8; ++r) {
            int i3 = ri + r + 8 * half;                 // C-layout: M = r + 8*(lane/16)
            s3[i3 * LP + cj + mrow] = fmaxf(acc[r] + bb3, 0.f);
        }
    }
    __syncthreads();

    // ---- conv4 along b, dil 2 (WMMA) + BN + residual ReLU + store ----------
    // out4 = out3(16x20) x K4(20x16), K4[k][n] = w4[t] iff k == n + 2t.
    if (wave < 4) {
        const int ri = (wave >> 1) * 16;
        const int cj = (wave & 1) * 16;
        v8f acc = {0.f,0.f,0.f,0.f,0.f,0.f,0.f,0.f};
        #pragma unroll
        for (int ck = 0; ck < 5; ++ck) {
            const int k0 = 4 * ck + 2 * half;
            v2f A = *(const v2f*)&s3[(ri + mrow) * LP + cj + k0];  // A-layout pair load
            v2f Bv;
            Bv.x = (k0     == mrow) ? w4a : (k0     == mrow + 2) ? w4b : (k0     == mrow + 4) ? w4c : 0.f;
            Bv.y = (k0 + 1 == mrow) ? w4a : (k0 + 1 == mrow + 2) ? w4b : (k0 + 1 == mrow + 4) ? w4c : 0.f;
            acc = __builtin_amdgcn_wmma_f32_16x16x4_f32(false, A, false, Bv,
                                                        (short)0, acc, false, false);
        }
        #pragma unroll
        for (int r = 0; r < 8; ++r) {
            int i4 = ri + r + 8 * half;
            int j4 = cj + mrow;
            float o = acc[r] + bb4;
            o = fmaf(o - mu, scale, beta);                         // BN
            float xin = s0[(i4 + HALO) * LP + (j4 + HALO)];        // residual input
            oplane[(ta + i4) * DB + (tb + j4)] = fmaxf(o + xin, 0.f);
        }
    }
}

// ---------------------------------------------------------------------------
// TDM-based tiled transpose. Output dims fixed (8,128,128,128), flat
// out[((j0*128+j1)*128+j2)*128+j3]. Input strides per OUTPUT coord: s0 for j0,
// sm for the non-unit middle coord, 1 for coord "r" (j1 if r_is_1 else j2),
// s3 for j3. TDM loads a 32(j3-rows, stride s3) x 32(contig) tile into LDS
// with 1-DWORD pad every 32 DWORDs (pitch 33 -> conflict-free column reads).
// ---------------------------------------------------------------------------
__global__ __launch_bounds__(256)
void transpose_tdm(const float* __restrict__ in, float* __restrict__ out,
                   long long s0, long long sm, long long s3, int r_is_1)
{
    __shared__ __attribute__((aligned(16))) float tile[32 * 33];

    const int blk = blockIdx.x;
    const int t3  = (blk & 3) * 32;          // j3 tile start
    const int trr = ((blk >> 2) & 3) * 32;   // r-coord tile start
    const int jm  = (blk >> 4) & 127;        // middle coord
    const int j0  = blk >> 11;

    if (threadIdx.x < 32) {
        const long long off = (long long)j0 * s0 + (long long)jm * sm
                            + (long long)trr + (long long)t3 * s3;
        const uint64_t ga   = (uint64_t)(const void*)(in + off);
        const unsigned lds  = (unsigned)(uintptr_t)&tile[0];

        u32x4 g0;
        g0.x = 1u;                                             // count=1
        g0.y = lds;                                            // lds_addr
        g0.z = (unsigned)(ga & 0xFFFFFFFFu);                   // global_addr lo
        g0.w = (unsigned)((ga >> 32) & 0x01FFFFFFu) | (2u << 30); // hi + type=2

        const unsigned td0 = 0x7FFFFFFFu, td1 = 0x7FFFFFFFu;   // no OOB clipping
        const uint64_t st0 = (uint64_t)s3;                     // row stride (elems)
        i32x8 g1;
        g1[0] = (int)((2u << 16) | (1u << 20) | (4u << 22));   // data_size=4B, pad 1dw/32dw
        g1[1] = (int)((td0 & 0xFFFFu) << 16);                  // tensor_dim0 lo16
        g1[2] = (int)(((td0 >> 16) & 0xFFFFu) | ((td1 & 0xFFFFu) << 16));
        g1[3] = (int)(((td1 >> 16) & 0xFFFFu) | (32u << 16));  // tile_dim0 = 32
        g1[4] = (int)32u;                                      // tile_dim1 = 32, tile_dim2 = 0
        g1[5] = (int)(unsigned)(st0 & 0xFFFFFFFFu);            // dim0_stride lo32
        g1[6] = (int)(unsigned)((st0 >> 32) & 0xFFFFu);        // dim0_stride hi16
        g1[7] = 0;

        i32x4 z4 = {0, 0, 0, 0};
#if defined(__clang_major__) && (__clang_major__ >= 23)
        i32x8 z8 = {0, 0, 0, 0, 0, 0, 0, 0};
        __builtin_amdgcn_tensor_load_to_lds(g0, g1, z4, z4, z8, 0);
#else
        __builtin_amdgcn_tensor_load_to_lds(g0, g1, z4, z4, 0);
#endif
        __builtin_amdgcn_s_wait_tensorcnt(0);
    }
    __syncthreads();

    const int y  = threadIdx.x & 31;    // j3 offset (lane-fast => coalesced stores)
    const int xb = threadIdx.x >> 5;
    const int j3 = t3 + y;
    for (int x = xb; x < 32; x += 8) {
        const int jr = trr + x;
        const int j1 = r_is_1 ? jr : jm;
        const int j2 = r_is_1 ? jm : jr;
        out[(((size_t)j0 * 128 + j1) * 128 + j2) * 128 + j3] = tile[y * 33 + x];
    }
}

// out(n,c,h,w) = in(n,h,c,w) : both sides coalesced along w, float4.
__global__ __launch_bounds__(256)
void swap12(const float* __restrict__ in, float* __restrict__ out)
{
    size_t i = (size_t)blockIdx.x * 256 + threadIdx.x;   // over 8*128*128*32 float4
    int w4 = (int)(i & 31);
    int h  = (int)((i >> 5) & 127);
    int c  = (int)((i >> 12) & 127);
    int n  = (int)(i >> 19);
    const float4* src = (const float4*)in  + ((((size_t)n * 128 + h) * 128 + c) * 32 + w4);
    float4*       dst = (float4*)out       + ((((size_t)n * 128 + c) * 128 + h) * 32 + w4);
    *dst = *src;
}

__global__ __launch_bounds__(256)
void copy4(const float4* __restrict__ in, float4* __restrict__ out)
{
    size_t i = (size_t)blockIdx.x * 256 + threadIdx.x;
    out[i] = in[i];
}

// ---------------------------------------------------------------------------
extern "C" void kernel_launch(void* const* d_in, const int* in_sizes, int n_in,
                              void* d_out, int out_size, void* d_ws, size_t ws_size,
                              hipStream_t stream)
{
    (void)in_sizes; (void)n_in; (void)out_size; (void)ws_size;

    const float* x = (const float*)d_in[0];
    const float* P[36];
    for (int i = 0; i < 36; ++i) P[i] = (const float*)d_in[1 + i];

    const size_t CHW = (size_t)NB * CHN * DA * DB;   // 16,777,216 elements
    float* out0 = (float*)d_out;
    float* out1 = out0 + CHW;
    float* out2 = out0 + 2 * CHW;
    float* out3 = out0 + 3 * CHW;
    float* ws_a = (float*)d_ws;
    float* ws_b = ws_a + CHW;

    const dim3 grid(16384), blk(256);

    // chunk0 = x
    copy4<<<grid, blk, 0, stream>>>((const float4*)x, (float4*)out0);

    // chunk1 = chw = pseudo(x, hw)
    pseudo_stage<<<grid, blk, 0, stream>>>(x, out1,
        P[0], P[1], P[2], P[3], P[4], P[5], P[6], P[7], P[8], P[9], P[10], P[11]);

    // ws_a = perm(chw,(0,3,1,2)) : out(n,w,c,h); strides(w)=1,(c)=16384,(h)=128
    transpose_tdm<<<grid, blk, 0, stream>>>(out1, ws_a, 2097152LL, 16384LL, 128LL, 1);

    // ws_b = wch = pseudo(ws_a, ch)   (layout n,W,C,H)
    pseudo_stage<<<grid, blk, 0, stream>>>(ws_a, ws_b,
        P[12], P[13], P[14], P[15], P[16], P[17], P[18], P[19], P[20], P[21], P[22], P[23]);

    // chunk2 = perm(wch,(0,2,3,1)) : out(n,c,h,w); strides(c)=128,(h)=1,(w)=16384
    transpose_tdm<<<grid, blk, 0, stream>>>(ws_b, out2, 2097152LL, 128LL, 16384LL, 0);

    // ws_a = perm(wch,(0,3,2,1)) : out(n,h,c,w); strides(h)=1,(c)=128,(w)=16384
    transpose_tdm<<<grid, blk, 0, stream>>>(ws_b, ws_a, 2097152LL, 128LL, 16384LL, 1);

    // ws_b = hcw = pseudo(ws_a, cw)   (layout n,H,C,W)
    pseudo_stage<<<grid, blk, 0, stream>>>(ws_a, ws_b,
        P[24], P[25], P[26], P[27], P[28], P[29], P[30], P[31], P[32], P[33], P[34], P[35]);

    // chunk3 = perm(hcw,(0,2,1,3))
    swap12<<<grid, blk, 0, stream>>>(ws_b, out3);
}